// NodeUpdateNetwork_42863773614333
// MI455X (gfx1250) — compile-verified
//
#include <hip/hip_runtime.h>

// B=8, N=128, T=512, F=64, NF=64
// Per (b,t) slice:
//   aggr(128x64) = Enorm(128x128) @ Node(128x64)
//   H(128x128)   = leaky([Node|aggr](128x128) @ W0^T)
//   Out(128x64)  = leaky(H @ W1^T)
// FP32 via V_WMMA_F32_16X16X4_F32; bulk HBM traffic via ASYNC global<->LDS ops.

typedef __attribute__((ext_vector_type(2))) float v2f;
typedef __attribute__((ext_vector_type(8))) float v8f;

#define LRELU_SLOPE 0.01f

#define WMMA_F32(a, b, c) \
    __builtin_amdgcn_wmma_f32_16x16x4_f32(false, (a), false, (b), (short)0, (c), false, false)

// 8-byte vector load (merges to ds_load_b64 / global_load_b64)
__device__ __forceinline__ v2f ld2(const float* p) { return *(const v2f*)p; }

// LDS byte address of a __shared__ object = low 32 bits of its generic address
__device__ __forceinline__ unsigned lds_addr_of(const void* p) {
    return (unsigned)(unsigned long long)p;
}

// Async copy 16B per lane: global -> LDS (ASYNCcnt-tracked)
__device__ __forceinline__ void async_g2l_b128(unsigned lds_byte, const void* gaddr) {
    asm volatile("global_load_async_to_lds_b128 %0, %1, off"
                 :: "v"(lds_byte), "v"(gaddr)
                 : "memory");
}

// Async copy 16B per lane: LDS -> global (ASYNCcnt-tracked)
__device__ __forceinline__ void async_l2g_b128(const void* gaddr, unsigned lds_byte) {
    asm volatile("global_store_async_from_lds_b128 %0, %1, off"
                 :: "v"(gaddr), "v"(lds_byte)
                 : "memory");
}

__device__ __forceinline__ void wait_async0() {
    asm volatile("s_wait_asynccnt 0x0" ::: "memory");
}

// C/D 16x16 f32: VGPR r -> lanes 0-15: M=m0+r, N=lane; lanes 16-31: M=m0+8+r, N=lane-16
__device__ __forceinline__ void storeC(float* M, int ld, int m0, int n0, int lane, v8f c) {
    float* p = M + (m0 + ((lane >> 4) << 3)) * ld + n0 + (lane & 15);
#pragma unroll
    for (int r = 0; r < 8; ++r) p[r * ld] = c[r];
}

__device__ __forceinline__ v8f leaky8(v8f x) {
#pragma unroll
    for (int r = 0; r < 8; ++r) {
        float v = x[r];
        x[r] = v > 0.0f ? v : LRELU_SLOPE * v;
    }
    return x;
}

// ---------------------------------------------------------------------------
// Kernel 1: masked + L1-row-normalized edge matrix. One wave per row of 128.
// ---------------------------------------------------------------------------
__global__ __launch_bounds__(256) void enorm_kernel(const float* __restrict__ edge,
                                                    float* __restrict__ enorm) {
    int row = blockIdx.x * 8 + (threadIdx.x >> 5);  // b*128 + i
    int lane = threadIdx.x & 31;
    int i = row & 127;

    const float4* e4 = (const float4*)(edge + (size_t)row * 128);
    float4 v = e4[lane];
    int jb = lane * 4;
    if (i >= jb && i < jb + 4) ((float*)&v)[i - jb] = 0.0f;  // mask diagonal

    float s = fabsf(v.x) + fabsf(v.y) + fabsf(v.z) + fabsf(v.w);
#pragma unroll
    for (int off = 16; off > 0; off >>= 1) s += __shfl_xor(s, off, 32);

    float inv = 1.0f / fmaxf(s, 1e-12f);
    float4 o;
    o.x = v.x * inv; o.y = v.y * inv; o.z = v.z * inv; o.w = v.w * inv;
    ((float4*)(enorm + (size_t)row * 128))[lane] = o;
}

// ---------------------------------------------------------------------------
// Kernel 2: fused 3-stage chain, one workgroup (8 wave32) per (b, t) slice.
// LDS: node 32K + nodeT 32K + aggr 32K + H 64K = 160 KB (2 WGs / 320 KB WGP).
// ---------------------------------------------------------------------------
__global__ __launch_bounds__(256) void fused_chain_kernel(const float* __restrict__ node,
                                                          const float* __restrict__ enorm,
                                                          const float* __restrict__ W0,
                                                          const float* __restrict__ W1,
                                                          float* __restrict__ out) {
    __shared__ __align__(16) float lds_node[128 * 64];    // node slice [n][f]
    __shared__ __align__(16) float lds_nodeT[64 * 128];   // transposed  [f][n] (stage-1 B frags)
    __shared__ __align__(16) float lds_aggr[128 * 64];    // aggr, later reused for output
    __shared__ __align__(16) float lds_h[128 * 128];      // hidden activations

    const int t = blockIdx.x;        // 0..511
    const int b = blockIdx.y;        // 0..7
    const int tid = threadIdx.x;
    const int lane = tid & 31;
    const int w = tid >> 5;          // wave id 0..7
    const int lrow = lane & 15;      // fragment M/N index within tile
    const int kh = (lane >> 4) << 1; // fragment K sub-offset {0,2}

    const int nrow = tid >> 1, half = tid & 1;  // row / 128-byte half assignment for copies
    const size_t slice_row = (((size_t)(b * 128 + nrow)) * 512 + t) * 64;

    // ---- Async copy node slice [128 x 64] global -> LDS (no VGPR staging) ----
    {
        const char* gsrc = (const char*)(node + slice_row) + half * 128;
        unsigned ldst = lds_addr_of(lds_node + nrow * 64) + half * 128;
#pragma unroll
        for (int i = 0; i < 8; ++i) async_g2l_b128(ldst + i * 16, gsrc + i * 16);
        wait_async0();               // this wave's LDS writes landed
    }
    __syncthreads();                 // publish across all 8 waves

    // ---- Build transposed copy nodeT[f][n] from LDS (b128 readback + scatter) ----
    {
        const float4* src = (const float4*)(lds_node + nrow * 64);
        float* tb = lds_nodeT + (half * 32) * 128 + nrow;
#pragma unroll
        for (int i = 0; i < 8; ++i) {
            float4 v = src[half * 8 + i];
            tb[(i * 4 + 0) * 128] = v.x;
            tb[(i * 4 + 1) * 128] = v.y;
            tb[(i * 4 + 2) * 128] = v.z;
            tb[(i * 4 + 3) * 128] = v.w;
        }
    }
    __syncthreads();

    // ---- Stage 1: aggr = Enorm @ node.  Wave w -> M-tile w ----
    {
        const float* Erow  = enorm + (size_t)b * 128 * 128 + (w * 16 + lrow) * 128 + kh;  // A (L2)
        const float* Bbase = lds_nodeT + lrow * 128 + kh;                                 // B (LDS)
        v8f zero = {};
        v8f acc[4] = {zero, zero, zero, zero};
#pragma unroll
        for (int j = 0; j < 32; ++j) {
            const int k0 = 4 * j;
            v2f a = ld2(Erow + k0);
#pragma unroll
            for (int nt = 0; nt < 4; ++nt) {
                v2f bf = ld2(Bbase + nt * 16 * 128 + k0);
                acc[nt] = WMMA_F32(a, bf, acc[nt]);
            }
        }
#pragma unroll
        for (int nt = 0; nt < 4; ++nt) storeC(lds_aggr, 64, w * 16, nt * 16, lane, acc[nt]);
    }
    __syncthreads();

    // ---- Stage 2: H = leaky(X @ W0^T), X = [node | aggr].  Wave w -> out-channel tile w ----
    {
        const float* An   = lds_node + lrow * 64 + kh;                 // A base, k in [0,64)
        const float* Aa   = lds_aggr + lrow * 64 + kh;                 // A base, k in [64,128)
        const float* Brow = W0 + (w * 16 + lrow) * 128 + kh;           // B base (L2)
        v8f zero = {};
        v8f acc[8] = {zero, zero, zero, zero, zero, zero, zero, zero};
#pragma unroll
        for (int j = 0; j < 32; ++j) {
            const int k0 = 4 * j;
            v2f bf = ld2(Brow + k0);
            const float* Xs = (k0 < 64) ? (An + k0) : (Aa + (k0 - 64));
#pragma unroll
            for (int mt = 0; mt < 8; ++mt) {
                v2f a = ld2(Xs + mt * 16 * 64);
                acc[mt] = WMMA_F32(a, bf, acc[mt]);
            }
        }
#pragma unroll
        for (int mt = 0; mt < 8; ++mt) storeC(lds_h, 128, mt * 16, w * 16, lane, leaky8(acc[mt]));
    }
    __syncthreads();

    // ---- Stage 3: Out = leaky(H @ W1^T).  Wave w -> n-tile (w&3), M rows (w>>2)*64.. ----
    {
        const int n0 = (w & 3) * 16;
        const int mbase = (w >> 2) * 64;
        const float* Ah   = lds_h + (mbase + lrow) * 128 + kh;
        const float* Brow = W1 + (n0 + lrow) * 128 + kh;               // B base (L2)
        v8f zero = {};
        v8f acc[4] = {zero, zero, zero, zero};
#pragma unroll
        for (int j = 0; j < 32; ++j) {
            const int k0 = 4 * j;
            v2f bf = ld2(Brow + k0);
#pragma unroll
            for (int mt = 0; mt < 4; ++mt) {
                v2f a = ld2(Ah + mt * 16 * 128 + k0);
                acc[mt] = WMMA_F32(a, bf, acc[mt]);
            }
        }
#pragma unroll
        for (int mt = 0; mt < 4; ++mt)
            storeC(lds_aggr, 64, mbase + mt * 16, n0, lane, leaky8(acc[mt]));  // reuse aggr buf
    }
    __syncthreads();                 // output slice complete in lds_aggr

    // ---- Async writeout LDS -> global (no VGPR round-trip) ----
    {
        const char* gdst = (const char*)(out + slice_row) + half * 128;
        unsigned lsrc = lds_addr_of(lds_aggr + nrow * 64) + half * 128;
#pragma unroll
        for (int i = 0; i < 8; ++i) async_l2g_b128(gdst + i * 16, lsrc + i * 16);
        wait_async0();               // drain before wave teardown
    }
}

extern "C" void kernel_launch(void* const* d_in, const int* in_sizes, int n_in,
                              void* d_out, int out_size, void* d_ws, size_t ws_size,
                              hipStream_t stream) {
    const float* node = (const float*)d_in[0];  // [8,128,512,64]
    const float* edge = (const float*)d_in[1];  // [8,128,128]
    const float* W0   = (const float*)d_in[2];  // [128,128]
    const float* W1   = (const float*)d_in[3];  // [64,128]
    float* out = (float*)d_out;                 // [8,128,512,64]
    float* enorm = (float*)d_ws;                // 8*128*128 floats = 512 KB scratch

    (void)in_sizes; (void)n_in; (void)out_size; (void)ws_size;

    enorm_kernel<<<128, 256, 0, stream>>>(edge, enorm);
    fused_chain_kernel<<<dim3(512, 8), 256, 0, stream>>>(node, enorm, W0, W1, out);
}